// Embedding_59768764891279
// MI455X (gfx1250) — compile-verified
//
#include <hip/hip_runtime.h>

// Embedding gather: out[token, :] = W[ids[token], :]
//   ids : int32  [8192]          (4 x 2048 tokens)
//   W   : float32[32000, 128]    = 16 MB (L2-resident on MI455X, 192 MB L2)
//   out : float32[8192, 128]     = 4 MB
//
// Pure memory-bound gather (~8-20 MB traffic -> sub-microsecond at 23.3 TB/s).
// One wave32 per token: lane l moves bytes [16l,16l+16) of the 512 B row.
// - id lookup scalarized via readfirstlane -> SGPR row base, lanes add lane*16.
// - all offsets 32-bit unsigned (id*128 <= 4.1M elements) -> saddr+voffset form.
// - NT stores: output is write-once stream, keep L2 for the weight table.
// - branchless global_prefetch_b8 of next iteration's row (clamped).

typedef __attribute__((ext_vector_type(4))) float f4;

#define EMBED_DIM     128u
#define BLOCK_THREADS 256   // 8 waves per block (wave32)
#define GRID_BLOCKS   256   // 2048 waves -> 4 tokens per wave at 8192 tokens

__global__ __launch_bounds__(BLOCK_THREADS) void embed_gather_kernel(
    const int* __restrict__ ids,
    const float* __restrict__ W,
    float* __restrict__ out,
    unsigned n_tokens,
    unsigned wave_stride) {
  const unsigned gid  = blockIdx.x * BLOCK_THREADS + threadIdx.x;
  const unsigned wave = gid >> 5;     // wave32: one token per wave per iteration
  const unsigned lane = gid & 31u;    // lane covers floats [4*lane, 4*lane+4)
  const unsigned last = n_tokens - 1u;

  for (unsigned token = wave; token < n_tokens; token += wave_stride) {
    // token is uniform across the wave by construction; make it explicit so the
    // id load and row base become scalar (SGPR) operations.
    const unsigned t  = __builtin_amdgcn_readfirstlane(token);
    const unsigned id = (unsigned)ids[t];          // wave-uniform scalar load

    // Branchless prefetch of the next iteration's row (clamped to last token;
    // re-prefetching a resident row is harmless). Speculative, no LOADcnt cost.
    const unsigned ntok = token + wave_stride;
    const unsigned nt   = __builtin_amdgcn_readfirstlane(ntok <= last ? ntok : last);
    const unsigned nid  = (unsigned)ids[nt];
    __builtin_prefetch(W + nid * EMBED_DIM + lane * 4u, /*rw=*/0, /*loc=*/3);

    // 512 B row: 32 lanes x 16 B -> one coalesced global_load_b128 per lane.
    const f4 v = ((const f4*)(W + id * EMBED_DIM))[lane];

    // Write-once stream -> non-temporal global_store_b128.
    __builtin_nontemporal_store(v, (f4*)(out + t * EMBED_DIM) + lane);
  }
}

extern "C" void kernel_launch(void* const* d_in, const int* in_sizes, int n_in,
                              void* d_out, int out_size, void* d_ws, size_t ws_size,
                              hipStream_t stream) {
  const int*   ids = (const int*)d_in[0];    // input_ids, int32, flat [8192]
  const float* W   = (const float*)d_in[1];  // weight, float32, [32000*128]
  float*       out = (float*)d_out;          // [8192*128] float32

  const unsigned n_tokens    = (unsigned)in_sizes[0];              // 8192
  const unsigned wave_stride = GRID_BLOCKS * (BLOCK_THREADS / 32); // 2048 waves

  embed_gather_kernel<<<GRID_BLOCKS, BLOCK_THREADS, 0, stream>>>(
      ids, W, out, n_tokens, wave_stride);
}